// GCNEncoder_82970178224660
// MI455X (gfx1250) — compile-verified
//
#include <hip/hip_runtime.h>

typedef __attribute__((ext_vector_type(2))) float v2f;
typedef __attribute__((ext_vector_type(4))) float v4f;
typedef __attribute__((ext_vector_type(8))) float v8f;

#define CH 128   // IN_CH == OUT_CH == 128

// ---------------- degree / normalization ----------------

__global__ void k_init_deg(float* __restrict__ deg, int n) {
    int i = blockIdx.x * blockDim.x + threadIdx.x;
    if (i < n) deg[i] = 1.0f;   // self-loop weight pre-added
}

__global__ void k_accum_deg(const long long* __restrict__ ei,
                            const float* __restrict__ ew,
                            float* __restrict__ deg, int E) {
    int e = blockIdx.x * blockDim.x + threadIdx.x;
    if (e < E) {
        long long col = ei[(long long)E + e];   // target node
        atomicAdd(&deg[col], ew[e]);
    }
}

__global__ void k_dinv(float* __restrict__ deg, int n) {
    int i = blockIdx.x * blockDim.x + threadIdx.x;
    if (i < n) deg[i] = rsqrtf(deg[i]);         // deg >= 1 always (self-loop)
}

// ---------------- fp32 WMMA GEMM: xw = x @ W; out = dinv^2 * xw ----------------
// One wave computes a 16-row strip across all 8 column tiles (16x16 each),
// K advanced 4 at a time with V_WMMA_F32_16X16X4_F32 (exact fp32 math).
// W is staged in LDS PRE-PAIRED in B-fragment layout:
//   Wp[p*CH + c] = { W[2p][c], W[2p+1][c] }   (p = k/2)
// so each B fragment is a single conflict-free ds_load_b64 into an even VGPR
// pair -- no v_mov reassembly in the inner loop.

__global__ __launch_bounds__(256) void k_gemm_wmma(
        const float* __restrict__ x, const float* __restrict__ W,
        const float* __restrict__ dinv,
        float* __restrict__ xw, float* __restrict__ out, int n) {
    __shared__ v2f Wp[(CH / 2) * CH];           // 64 KB
    float* Wpf = (float*)Wp;
    for (int f = threadIdx.x; f < CH * CH; f += 256) {
        int k = f >> 7, c = f & (CH - 1);
        Wpf[((k >> 1) * CH + c) * 2 + (k & 1)] = W[f];
    }
    __syncthreads();

    int wid  = (blockIdx.x * 256 + threadIdx.x) >> 5;   // global wave id = row tile
    int lane = threadIdx.x & 31;
    int row0 = wid * 16;
    if (row0 >= n) return;

    int m  = lane & 15;          // M (for A) / N (for B) index within tile
    int hk = lane >> 4;          // half select: lanes 0-15 -> K 0,1 ; 16-31 -> K 2,3
    int kh = hk << 1;

    v8f acc[8] = {};             // 8 column tiles of 16x16 f32

    int rload = row0 + m; if (rload > n - 1) rload = n - 1;   // tail clamp (N%16==0 normally)
    const float* xrow = x + (long long)rload * CH;
    const v2f* wrow = Wp + m;    // lane's column within pair rows

    for (int k = 0; k < CH; k += 4) {
        // A fragment: 16x4 tile of x  (ISA layout: vgpr0=K+kh, vgpr1=K+kh+1)
        v2f a = *(const v2f*)(xrow + k + kh);
        int p = (k >> 1) + hk;   // pair-row index for this lane's K half
        #pragma unroll
        for (int nt = 0; nt < 8; ++nt) {
            v2f bm = wrow[p * CH + nt * 16];   // single ds_load_b64
            acc[nt] = __builtin_amdgcn_wmma_f32_16x16x4_f32(
                false, a, false, bm, (short)0, acc[nt], false, false);
        }
    }

    // C/D layout: vgpr v -> row v + 8*hk, column = lane&15 (+ tile offset)
    #pragma unroll
    for (int v = 0; v < 8; ++v) {
        int rr = row0 + v + (hk << 3);
        if (rr < n) {
            float di = dinv[rr];
            float d2 = di * di;                 // self-loop norm
            long long base = (long long)rr * CH + m;
            #pragma unroll
            for (int nt = 0; nt < 8; ++nt) {
                float val = acc[nt][v];
                xw[base + nt * 16]  = val;
                out[base + nt * 16] = val * d2; // seed accumulator w/ self-loop term
            }
        }
    }
}

// ---------------- edge scatter: one wave per edge ----------------
// 32 lanes x float4 = 128 channels. xw/out are L2-resident (51 MB each vs 192 MB L2).

__global__ __launch_bounds__(256) void k_scatter(
        const long long* __restrict__ ei, const float* __restrict__ ew,
        const float* __restrict__ dinv, const float* __restrict__ xw,
        float* __restrict__ out, int E) {
    long long gid = (long long)blockIdx.x * 256 + threadIdx.x;
    int e    = (int)(gid >> 5);
    int lane = (int)(gid & 31);
    if (e >= E) return;

    long long r = ei[e];
    long long c = ei[(long long)E + e];
    float nrm = dinv[r] * ew[e] * dinv[c];

    v4f v = *(const v4f*)(xw + r * CH + lane * 4);
    float* dst = out + c * CH + lane * 4;
    atomicAdd(dst + 0, v.x * nrm);
    atomicAdd(dst + 1, v.y * nrm);
    atomicAdd(dst + 2, v.z * nrm);
    atomicAdd(dst + 3, v.w * nrm);
}

// ---------------- bias + relu ----------------

__global__ void k_bias_relu(float* __restrict__ out, const float* __restrict__ b, int total) {
    int i = blockIdx.x * blockDim.x + threadIdx.x;
    if (i < total) out[i] = fmaxf(out[i] + b[i & (CH - 1)], 0.0f);
}

// ---------------- launch ----------------

extern "C" void kernel_launch(void* const* d_in, const int* in_sizes, int n_in,
                              void* d_out, int out_size, void* d_ws, size_t ws_size,
                              hipStream_t stream) {
    const float*     x  = (const float*)d_in[0];
    const long long* ei = (const long long*)d_in[1];   // int64 [2,E]
    const float*     ew = (const float*)d_in[2];
    const float*     W  = (const float*)d_in[3];
    const float*     b  = (const float*)d_in[4];
    float*           out = (float*)d_out;

    int n = in_sizes[0] / CH;   // 100000
    int E = in_sizes[2];        // 1600000

    float* deg = (float*)d_ws;      // n floats, becomes dinv in place
    float* xw  = deg + n;           // n*CH floats

    k_init_deg <<<(n + 255) / 256, 256, 0, stream>>>(deg, n);
    k_accum_deg<<<(E + 255) / 256, 256, 0, stream>>>(ei, ew, deg, E);
    k_dinv     <<<(n + 255) / 256, 256, 0, stream>>>(deg, n);

    int rowTiles   = (n + 15) / 16;          // one wave per 16-row strip
    int gemmBlocks = (rowTiles + 7) / 8;     // 8 waves (256 thr) per block
    k_gemm_wmma<<<gemmBlocks, 256, 0, stream>>>(x, W, deg, xw, out, n);

    long long sthreads = (long long)E * 32;  // one wave per edge
    k_scatter<<<(int)((sthreads + 255) / 256), 256, 0, stream>>>(ei, ew, deg, xw, out, E);

    int total = n * CH;
    k_bias_relu<<<(total + 255) / 256, 256, 0, stream>>>(out, b, total);
}